// FeatureDecoder_5162550690550
// MI455X (gfx1250) — compile-verified
//
#include <hip/hip_runtime.h>
#include <math.h>

typedef __attribute__((ext_vector_type(2))) float v2f;
typedef __attribute__((ext_vector_type(8))) float v8f;

#define BATCH 2
#define NQ    100        // K dimension (divisible by 4 -> 25 WMMA steps)
#define NC    151
#define CPAD  160        // C padded to 10 tiles of 16
#define IMH   128
#define IMW   128
#define HW    16384      // 128*128
#define NCOUT 150        // channels 1..150 kept
#define OUTW  512

// ---------------- workspace layout (floats) ----------------
// [0, 3276800)            sigmoid(masks)   B*NQ*HW
// [3276800, 3308800)      scoresT          B*CPAD*NQ   (A matrix, transposed, zero-padded)
// [3308800, 8224000)      seg              B*NCOUT*HW  (pre-resize einsum result)
#define SIGM_OFF   0
#define SCORE_OFF  3276800
#define SEG_OFF    3308800

// -------- elementwise sigmoid of pred_masks --------
__global__ void k_sigmoid(const float* __restrict__ in, float* __restrict__ out, int n) {
    int i = blockIdx.x * blockDim.x + threadIdx.x;
    if (i < n) {
        float x = in[i];
        out[i] = 1.0f / (1.0f + __expf(-x));
    }
}

// -------- softmax over C per (b,q), written transposed [b][c][q], c padded to CPAD --------
__global__ void k_softmax_T(const float* __restrict__ logits, float* __restrict__ scoresT) {
    int row  = blockIdx.x * (blockDim.x >> 5) + (threadIdx.x >> 5);   // b*NQ + q
    int lane = threadIdx.x & 31;
    if (row >= BATCH * NQ) return;

    const float* src = logits + (size_t)row * NC;
    float vals[5];
    float m = -INFINITY;
    #pragma unroll
    for (int i = 0; i < 5; i++) {
        int c = lane + 32 * i;
        float v = (c < NC) ? src[c] : -INFINITY;
        vals[i] = v;
        m = fmaxf(m, v);
    }
    #pragma unroll
    for (int off = 16; off > 0; off >>= 1) m = fmaxf(m, __shfl_xor(m, off, 32));

    float s = 0.0f;
    #pragma unroll
    for (int i = 0; i < 5; i++) {
        int c = lane + 32 * i;
        if (c < NC) { vals[i] = __expf(vals[i] - m); s += vals[i]; }
    }
    #pragma unroll
    for (int off = 16; off > 0; off >>= 1) s += __shfl_xor(s, off, 32);
    float inv = 1.0f / s;

    int b = row / NQ, q = row % NQ;
    float* dst = scoresT + (size_t)b * CPAD * NQ + q;
    #pragma unroll
    for (int i = 0; i < 5; i++) {
        int c = lane + 32 * i;                       // up to 159
        dst[(size_t)c * NQ] = (c < NC) ? vals[i] * inv : 0.0f;
    }
}

// -------- WMMA f32 GEMM: seg[b][c-1][p] = sum_q scoresT[b][c][q] * sigm[b][q][p] --------
// One wave per 16(M=c) x 16(N=p) tile; K=100 = 25 steps of V_WMMA_F32_16X16X4_F32.
__global__ void k_gemm_wmma(const float* __restrict__ scoresT,
                            const float* __restrict__ sigm,
                            float* __restrict__ seg) {
    int wave = (blockIdx.x * blockDim.x + threadIdx.x) >> 5;
    int lane = threadIdx.x & 31;
    const int CT = CPAD / 16;   // 10
    const int PT = HW / 16;     // 1024

    int b  = wave / (CT * PT);
    int r  = wave % (CT * PT);
    int ct = r / PT;
    int pt = r % PT;

    int mn = lane & 15;         // M row (for A) == N col (for B/D) per ISA layout
    int hi = lane >> 4;         // half-wave select: K pair {0,1} vs {2,3}

    const float* Arow  = scoresT + (size_t)(b * CPAD + ct * 16 + mn) * NQ;   // + k
    const float* Bbase = sigm + (size_t)b * NQ * HW + pt * 16 + mn;          // + k*HW

    v8f acc = {};
    #pragma unroll 5
    for (int k0 = 0; k0 < NQ; k0 += 4) {
        int k = k0 + 2 * hi;
        v2f a;  a.x  = Arow[k];                 a.y  = Arow[k + 1];
        v2f bb; bb.x = Bbase[(size_t)k * HW];   bb.y = Bbase[(size_t)(k + 1) * HW];
        acc = __builtin_amdgcn_wmma_f32_16x16x4_f32(
                  /*neg_a=*/false, a, /*neg_b=*/false, bb,
                  /*c_mod=*/(short)0, acc, /*reuse_a=*/false, /*reuse_b=*/false);
    }

    // D layout: VGPR rr, lanes0-15 -> M=rr, lanes16-31 -> M=rr+8. Keep channels 1..150.
    #pragma unroll
    for (int rr = 0; rr < 8; rr++) {
        int c = ct * 16 + rr + 8 * hi;
        if (c >= 1 && c <= NCOUT) {
            seg[(size_t)(b * NCOUT + (c - 1)) * HW + pt * 16 + mn] = acc[rr];
        }
    }
}

// -------- bilinear 128 -> 512 upsample (half-pixel centers, edge clamp) --------
// Each thread produces 4 consecutive x outputs and issues one float4 store.
__global__ void k_resize(const float* __restrict__ seg, float* __restrict__ out) {
    long idx = (long)blockIdx.x * blockDim.x + threadIdx.x;
    const long total4 = (long)BATCH * NCOUT * OUTW * OUTW / 4;
    if (idx >= total4) return;

    long o  = idx * 4;
    int  x  = (int)(o % OUTW);
    long t  = o / OUTW;
    int  y  = (int)(t % OUTW);
    long bc = t / OUTW;                         // b*NCOUT + c

    const float* plane = seg + bc * (long)HW;

    float sy = (y + 0.5f) * 0.25f - 0.5f;
    float fy = floorf(sy);
    float wy = sy - fy;
    int y0 = (int)fy;
    int y0c = y0 < 0 ? 0 : y0;
    int y1c = (y0 + 1 > IMH - 1) ? (IMH - 1) : (y0 + 1);
    const float* row0 = plane + (size_t)y0c * IMW;
    const float* row1 = plane + (size_t)y1c * IMW;

    float4 res;
    float* rp = (float*)&res;
    #pragma unroll
    for (int i = 0; i < 4; i++) {
        float sx = (x + i + 0.5f) * 0.25f - 0.5f;
        float fx = floorf(sx);
        float wx = sx - fx;
        int x0 = (int)fx;
        int x0c = x0 < 0 ? 0 : x0;
        int x1c = (x0 + 1 > IMW - 1) ? (IMW - 1) : (x0 + 1);
        float v00 = row0[x0c], v01 = row0[x1c];
        float v10 = row1[x0c], v11 = row1[x1c];
        float top = v00 + wx * (v01 - v00);
        float bot = v10 + wx * (v11 - v10);
        rp[i] = top + wy * (bot - top);
    }
    *(float4*)(out + o) = res;
}

extern "C" void kernel_launch(void* const* d_in, const int* in_sizes, int n_in,
                              void* d_out, int out_size, void* d_ws, size_t ws_size,
                              hipStream_t stream) {
    const float* logits = (const float*)d_in[0];   // (2,100,151) f32
    const float* masks  = (const float*)d_in[1];   // (2,100,128,128) f32
    float* ws      = (float*)d_ws;
    float* sigm    = ws + SIGM_OFF;
    float* scoresT = ws + SCORE_OFF;
    float* seg     = ws + SEG_OFF;
    float* outp    = (float*)d_out;                // (2,150,512,512) f32

    // 1) sigmoid(masks) -> ws
    int nSig = BATCH * NQ * HW;                    // 3,276,800
    k_sigmoid<<<(nSig + 255) / 256, 256, 0, stream>>>(masks, sigm, nSig);

    // 2) softmax(logits) transposed + padded -> ws
    k_softmax_T<<<(BATCH * NQ + 7) / 8, 256, 0, stream>>>(logits, scoresT);

    // 3) WMMA einsum -> seg (exact wave coverage: 20480 waves, 8 waves/block)
    int nTiles = BATCH * (CPAD / 16) * (HW / 16);  // 20480
    k_gemm_wmma<<<nTiles / 8, 256, 0, stream>>>(scoresT, sigm, seg);

    // 4) bilinear upsample -> d_out (write-bound: 314.6 MB)
    long total4 = (long)BATCH * NCOUT * OUTW * OUTW / 4;   // 19,660,800
    k_resize<<<(int)((total4 + 255) / 256), 256, 0, stream>>>(seg, outp);
}